// Gate_82626580841192
// MI455X (gfx1250) — compile-verified
//
#include <hip/hip_runtime.h>
#include <hip/hip_bf16.h>
#include <stdint.h>

// MoE gate: scores = sigmoid(x @ W^T + b); grouped top-k routing (DeepSeek style).
// N=16384 tokens, D=2048, E=64 experts, G=8 groups, TOP_K=8, TOP_K_GROUP=4.
//
// GEMM via V_WMMA_F32_16X16X4_F32 (fp32 WMMA, preserves fp32 top-k ranking).
// Data movement via Tensor Data Mover (TENSOR_LOAD_TO_LDS) with double-buffered
// K-chunks: wave0 issues chunk c+1 DMA, s_wait_tensorcnt(2), barrier, compute.
// TDM hardware padding (pad_interval=32 dw, pad_amount=4 dw) gives LDS row
// stride 36 floats -> bank-conflict-free b64 fragment loads (36r mod 64 = 4*(9r mod 16)).

typedef __attribute__((ext_vector_type(2))) float        v2f;
typedef __attribute__((ext_vector_type(8))) float        v8f;
typedef __attribute__((ext_vector_type(4))) unsigned int v4u;
typedef __attribute__((ext_vector_type(8))) int          v8i;
typedef __attribute__((ext_vector_type(4))) int          v4i;

#define DMODEL   2048
#define NEXP     64
#define KC       32                  // K chunk (elements)
#define LSTR     36                  // padded LDS row stride (floats) = KC + 4
#define SSTR     68                  // score row stride for epilogue transpose
#define TOKS_WG  64                  // tokens per workgroup (4 waves x 16)
#define NCHUNK   (DMODEL / KC)
#define NEG_INF  (-3.402823466e38f)

#if defined(__has_builtin)
#if __has_builtin(__builtin_amdgcn_tensor_load_to_lds)
#define HAVE_TDM 1
#endif
#endif
#ifndef HAVE_TDM
#define HAVE_TDM 0
#endif

#if HAVE_TDM
// 2-D tile DMA: global (row-major, rowStride elems of 4B) -> LDS at lds_off,
// tileW elems x tileH rows, LDS padding: +4 dwords every 32 dwords (stride 36).
// Descriptor bit layout per CDNA5 ISA sec 8.3 (group0) / 8.4 (group1).
static __device__ __forceinline__ void tdm_load_2d(uint32_t lds_off,
                                                   const float* gptr,
                                                   uint32_t tileW, uint32_t tileH,
                                                   uint32_t rowStride) {
    const uint64_t ga = (uint64_t)(uintptr_t)gptr;
    v4u g0;
    g0[0] = 1u;                                   // count=1 (valid), user mode
    g0[1] = lds_off;                              // lds_addr (bytes)
    g0[2] = (uint32_t)(ga & 0xFFFFFFFFu);         // global_addr[31:0]
    g0[3] = (uint32_t)((ga >> 32) & 0x01FFFFFFu)  // global_addr[56:32]
          | (2u << 30);                           // type=2 ("image")
    const uint32_t tdim0 = rowStride;             // generous in-bounds dims
    const uint32_t tdim1 = 1u << 20;
    v8i g1;
    g1[0] = (int)((2u << 16)                      // data_size: 4 bytes
                | (1u << 20)                      // pad_enable
                | (4u << 22)                      // pad_interval: code 4 = 32 dwords
                | (3u << 25));                    // pad_amount: code 3 = 4 dwords
    g1[1] = (int)((tdim0 & 0xFFFFu) << 16);       // [47:32]=atomic addr 0; [63:48]=tdim0 lo
    g1[2] = (int)((tdim0 >> 16) | ((tdim1 & 0xFFFFu) << 16));
    g1[3] = (int)((tdim1 >> 16) | (tileW << 16)); // [127:112] tile_dim0
    g1[4] = (int)(tileH & 0xFFFFu);               // tile_dim1; tile_dim2=0
    g1[5] = (int)rowStride;                       // tensor_dim0_stride lo32
    g1[6] = 0;                                    // stride hi16 | dim1_stride lo16
    g1[7] = 0;
    const v4i z4 = {0, 0, 0, 0};
#if defined(__clang_major__) && (__clang_major__ >= 23)
    const v8i z8 = {0, 0, 0, 0, 0, 0, 0, 0};
    __builtin_amdgcn_tensor_load_to_lds(g0, g1, z4, z4, z8, 0);
#else
    __builtin_amdgcn_tensor_load_to_lds(g0, g1, z4, z4, 0);
#endif
}
#endif  // HAVE_TDM

__global__ __launch_bounds__(128)
void Gate_82626580841192_kernel(const float* __restrict__ x,
                                const float* __restrict__ W,
                                const float* __restrict__ bvec,
                                const float* __restrict__ biasvec,
                                float* __restrict__ outW,
                                int*   __restrict__ outI) {
    __shared__ float xs[2 * TOKS_WG * LSTR];   // double-buffered x chunks
    __shared__ float ws[2 * NEXP   * LSTR];    // double-buffered W chunks
    __shared__ float bSh[NEXP];                // Linear bias (pre-sigmoid)
    __shared__ float biasSh[NEXP];             // routing bias (post-sigmoid)

    const int tid  = threadIdx.x;
    const int wave = tid >> 5;
    const int lane = tid & 31;
    const int l16  = lane & 15;
    const int koffLane = (lane >> 4) * 2;      // lanes 16..31 take K+2,K+3 halves
    const int rowA = wave * 16 + l16;          // A-fragment row for this lane

    if (tid < 64)       bSh[tid]         = bvec[tid];
    else                biasSh[tid - 64] = biasvec[tid - 64];

    const float* xg = x + (size_t)blockIdx.x * TOKS_WG * DMODEL;

#if HAVE_TDM
    const uint32_t xs_base = (uint32_t)(uintptr_t)&xs[0];
    const uint32_t ws_base = (uint32_t)(uintptr_t)&ws[0];
    const uint32_t xbufB   = TOKS_WG * LSTR * 4;   // bytes per x buffer
    const uint32_t wbufB   = NEXP   * LSTR * 4;    // bytes per W buffer
    if (wave == 0) {                                // prologue: chunk 0 -> buf 0
        tdm_load_2d(xs_base, xg, KC, TOKS_WG, DMODEL);
        tdm_load_2d(ws_base, W,  KC, NEXP,    DMODEL);
    }
#endif

    v8f acc[4] = {};   // 16(tok) x 64(exp) fp32 accumulators

    for (int c = 0; c < NCHUNK; ++c) {
        const int cur = c & 1;
#if HAVE_TDM
        if (wave == 0) {
            if (c + 1 < NCHUNK) {                  // issue-ahead chunk c+1
                const int nxt = cur ^ 1;
                const int k1  = (c + 1) * KC;
                tdm_load_2d(xs_base + (uint32_t)nxt * xbufB, xg + k1, KC, TOKS_WG, DMODEL);
                tdm_load_2d(ws_base + (uint32_t)nxt * wbufB, W  + k1, KC, NEXP,    DMODEL);
                __builtin_amdgcn_s_wait_tensorcnt(2);  // chunk c complete (in-order)
            } else {
                __builtin_amdgcn_s_wait_tensorcnt(0);
            }
        }
#else
        {   // fallback: VMEM float4 staging (coalesced)
            #pragma unroll
            for (int i = tid; i < TOKS_WG * (KC / 4); i += 128) {
                const int r = i >> 3, c4 = i & 7;
                float4 v = *(const float4*)(xg + (size_t)r * DMODEL + c * KC + c4 * 4);
                *(float4*)(&xs[(cur * TOKS_WG + r) * LSTR + c4 * 4]) = v;
            }
            #pragma unroll
            for (int i = tid; i < NEXP * (KC / 4); i += 128) {
                const int r = i >> 3, c4 = i & 7;
                float4 v = *(const float4*)(W + (size_t)r * DMODEL + c * KC + c4 * 4);
                *(float4*)(&ws[(cur * NEXP + r) * LSTR + c4 * 4]) = v;
            }
        }
#endif
        __syncthreads();

        const float* xb = &xs[cur * TOKS_WG * LSTR];
        const float* wb = &ws[cur * NEXP   * LSTR];
        #pragma unroll
        for (int kk = 0; kk < KC; kk += 4) {
            const int ko = kk + koffLane;
            // A 16x4 f32: lane<16 -> row l16 K=kk,kk+1 ; lane>=16 -> K=kk+2,kk+3
            v2f a  = *(const v2f*)(&xb[rowA * LSTR + ko]);
            v2f b0 = *(const v2f*)(&wb[( 0 + l16) * LSTR + ko]);
            v2f b1 = *(const v2f*)(&wb[(16 + l16) * LSTR + ko]);
            v2f b2 = *(const v2f*)(&wb[(32 + l16) * LSTR + ko]);
            v2f b3 = *(const v2f*)(&wb[(48 + l16) * LSTR + ko]);
            acc[0] = __builtin_amdgcn_wmma_f32_16x16x4_f32(false, a, false, b0,
                         (short)0, acc[0], false, false);
            acc[1] = __builtin_amdgcn_wmma_f32_16x16x4_f32(false, a, false, b1,
                         (short)0, acc[1], false, false);
            acc[2] = __builtin_amdgcn_wmma_f32_16x16x4_f32(false, a, false, b2,
                         (short)0, acc[2], false, false);
            acc[3] = __builtin_amdgcn_wmma_f32_16x16x4_f32(false, a, false, b3,
                         (short)0, acc[3], false, false);
        }
        __syncthreads();
    }

    // ---- transpose scores through LDS (reuse xs; 4*16*68=4352 <= 4608 floats) ----
    // C/D layout: VGPR r = row M=r (lanes 0-15) or M=r+8 (lanes 16-31), lane = N.
    float* sb = &xs[wave * 16 * SSTR];
    const int half8 = (lane >> 4) * 8;
    #pragma unroll
    for (int nt = 0; nt < 4; ++nt) {
        const int e  = nt * 16 + l16;
        const float bb = bSh[e];
        #pragma unroll
        for (int r = 0; r < 8; ++r) {
            const float t = acc[nt][r] + bb;
            sb[(r + half8) * SSTR + e] = 1.0f / (1.0f + __expf(-t));  // sigmoid
        }
    }
    __syncthreads();

    // ---- routing: one lane per token (lanes 0..15 of each wave) ----
    if (lane < 16) {
        const float* s = &xs[(wave * 16 + lane) * SSTR];  // sigmoid weights [64]

        // group score = sum of top-2 (score = sigmoid + routing bias) per group
        float gs[8];
        #pragma unroll
        for (int g = 0; g < 8; ++g) {
            float m1 = NEG_INF, m2 = NEG_INF;
            #pragma unroll
            for (int j = 0; j < 8; ++j) {
                const float v = s[g * 8 + j] + biasSh[g * 8 + j];
                if (v > m1) { m2 = m1; m1 = v; } else if (v > m2) { m2 = v; }
            }
            gs[g] = m1 + m2;
        }
        // top-4 groups (strict > == jax lowest-index tie-break)
        unsigned gmask = 0;
        #pragma unroll
        for (int k = 0; k < 4; ++k) {
            float best = NEG_INF; int bi = 0;
            #pragma unroll
            for (int g = 0; g < 8; ++g)
                if (!((gmask >> g) & 1) && gs[g] > best) { best = gs[g]; bi = g; }
            gmask |= 1u << bi;
        }
        // top-8 experts over masked scores (masked entries == 0.0, as reference)
        unsigned long long taken = 0ull;
        float wsel[8]; int isel[8];
        #pragma unroll
        for (int k = 0; k < 8; ++k) {
            float best = NEG_INF; int bi = 0;
            for (int e = 0; e < 64; ++e) {
                if ((taken >> e) & 1ull) continue;
                const bool on = (gmask >> (e >> 3)) & 1;
                const float sv = on ? (s[e] + biasSh[e]) : 0.0f;
                if (sv > best) { best = sv; bi = e; }
            }
            taken |= 1ull << bi;
            isel[k] = bi;
            wsel[k] = ((gmask >> (bi >> 3)) & 1) ? s[bi] : 0.0f;  // w_m gather
        }
        const int token = blockIdx.x * TOKS_WG + wave * 16 + lane;
        #pragma unroll
        for (int k = 0; k < 8; ++k) {
            outW[token * 8 + k] = wsel[k];
            outI[token * 8 + k] = isel[k];
        }
    }
}

extern "C" void kernel_launch(void* const* d_in, const int* in_sizes, int n_in,
                              void* d_out, int out_size, void* d_ws, size_t ws_size,
                              hipStream_t stream) {
    const float* x    = (const float*)d_in[0];
    const float* W    = (const float*)d_in[1];
    const float* b    = (const float*)d_in[2];
    const float* bias = (const float*)d_in[3];

    const int E = in_sizes[2];                 // 64
    const int D = in_sizes[1] / E;             // 2048
    const int N = in_sizes[0] / D;             // 16384

    float* outW = (float*)d_out;
    int*   outI = (int*)d_out + (size_t)N * 8; // idx stored as int32 after weights

    dim3 grid(N / TOKS_WG), block(128);
    Gate_82626580841192_kernel<<<grid, block, 0, stream>>>(x, W, b, bias, outW, outI);
}